// DADPolicy_36180804502161
// MI455X (gfx1250) — compile-verified
//
#include <hip/hip_runtime.h>

// ---------------------------------------------------------------------------
// CDNA5 (gfx1250) implementation.
//  - Encoder GEMMs (E x 5 @ 5 x 64, N x 2 @ 2 x 64) via V_WMMA_F32_16X16X4_F32
//    (mask-free main loop; tail tile handled once, masked)
//  - LSTM: single 256-thread block, one thread per gate row, h/c in LDS
//  - Fuse layer: ctx @ W1[:192] hoisted out of the 1M-candidate loop
// ---------------------------------------------------------------------------

typedef float v2f __attribute__((ext_vector_type(2)));
typedef float v8f __attribute__((ext_vector_type(8)));
typedef int   v2i __attribute__((ext_vector_type(2)));

#define NODE_BLOCKS 4
#define EDGE_BLOCKS 512
#define ENC_BLOCKS  (NODE_BLOCKS + EDGE_BLOCKS)
#define WPB 8   // waves per 256-thread block (wave32)

__device__ __forceinline__ v8f wmma4(v2f a, v2f b, v8f c) {
  // D(16x16,f32) = A(16x4,f32) @ B(4x16,f32) + C
  return __builtin_amdgcn_wmma_f32_16x16x4_f32(false, a, false, b, (short)0, c,
                                               false, false);
}

// single-instruction ReLU: med3(x, 0, +inf)
__device__ __forceinline__ float relu1(float x) {
  return __builtin_amdgcn_fmed3f(x, 0.f, __builtin_inff());
}

// ws layout (floats):
//  [0..63]    node column sums
//  [64..127]  edge column sums
//  [128..191] z_hist (LSTM final h)
//  [192..255] base   = fuse_b1 + ctx @ fuse_W1[:192]
//  [256..319] wa     = fuse_W1[192]
//  [320..383] wb     = fuse_W1[193]
//  [384..447] w2     = fuse_W2
//  [448]      b2

__global__ void zero_kernel(float* __restrict__ ws) {
  const int t = threadIdx.x;
  if (t < 128) ws[t] = 0.f;
}

__global__ __launch_bounds__(256) void encoder_kernel(
    const float* __restrict__ node, long node_rows,
    const float* __restrict__ edge, long edge_rows,
    const float* __restrict__ nodeW, const float* __restrict__ nodeB,
    const float* __restrict__ edgeW, const float* __restrict__ edgeB,
    float* __restrict__ ws)
{
  __shared__ float bsum[64];
  const int tid = threadIdx.x;
  if (tid < 64) bsum[tid] = 0.f;
  __syncthreads();

  const int lane = tid & 31;
  const int wid  = tid >> 5;
  const int half = lane >> 4;   // 0: K0/K1 lanes, 1: K2/K3 lanes
  const int l16  = lane & 15;

  v8f acc[4];
  #pragma unroll
  for (int t = 0; t < 4; ++t)
    #pragma unroll
    for (int r = 0; r < 8; ++r) acc[t][r] = 0.f;

  float* gout;

  if ((int)blockIdx.x < NODE_BLOCKS) {
    // ---------------- node encoder: K = 2 (pad to 4) ----------------
    v2f B[4]; v8f cb[4];
    #pragma unroll
    for (int t = 0; t < 4; ++t) {
      const int n = l16 + 16 * t;
      // B layout: lane l, reg r -> B[K = r + 2*half][N = l%16]
      B[t].x = half ? 0.f : nodeW[0 * 64 + n];   // K0 (K2 -> 0)
      B[t].y = half ? 0.f : nodeW[1 * 64 + n];   // K1 (K3 -> 0)
      const float bv = nodeB[n];                 // C[m][n] = b[n]
      #pragma unroll
      for (int r = 0; r < 8; ++r) cb[t][r] = bv;
    }
    const long full_tiles = node_rows >> 4;           // mask-free tiles
    const long stride     = (long)NODE_BLOCKS * WPB;
    const long w0         = (long)blockIdx.x * WPB + wid;

    for (long tile = w0; tile < full_tiles; tile += stride) {
      const float* p = node + (tile * 16 + l16) * 2;
      v2f a; a.x = 0.f; a.y = 0.f;
      if (half == 0) {
        a.x = __builtin_nontemporal_load(p + 0);
        a.y = __builtin_nontemporal_load(p + 1);
      }
      #pragma unroll
      for (int t = 0; t < 4; ++t) {
        v8f d = wmma4(a, B[t], cb[t]);
        #pragma unroll
        for (int r = 0; r < 8; ++r) acc[t][r] += relu1(d[r]);
      }
    }
    // masked tail tile (at most one per grid)
    if ((node_rows & 15) && (full_tiles % stride) == w0) {
      const long tile = full_tiles;
      const long row  = tile * 16 + l16;
      v2f a; a.x = 0.f; a.y = 0.f;
      if (half == 0 && row < node_rows) {
        a.x = node[row * 2];
        a.y = node[row * 2 + 1];
      }
      #pragma unroll
      for (int t = 0; t < 4; ++t) {
        v8f d = wmma4(a, B[t], cb[t]);
        #pragma unroll
        for (int r = 0; r < 8; ++r) {
          const long m = tile * 16 + r + 8 * half;
          acc[t][r] += (m < node_rows) ? relu1(d[r]) : 0.f;
        }
      }
    }
    gout = ws;          // node colsum -> ws[0..63]
  } else {
    // ---------------- edge encoder: K = 5 (K=4 WMMA + K=1 padded WMMA) -----
    v2f B03[4], B4[4]; v8f cb[4];
    #pragma unroll
    for (int t = 0; t < 4; ++t) {
      const int n = l16 + 16 * t;
      B03[t].x = edgeW[(2 * half + 0) * 64 + n];  // K = 2*half
      B03[t].y = edgeW[(2 * half + 1) * 64 + n];  // K = 2*half+1
      B4[t].x  = half ? 0.f : edgeW[4 * 64 + n];  // K0 row = W[4], rest 0
      B4[t].y  = 0.f;
      const float bv = edgeB[n];
      #pragma unroll
      for (int r = 0; r < 8; ++r) cb[t][r] = bv;
    }
    const long full_tiles = edge_rows >> 4;
    const long stride     = (long)EDGE_BLOCKS * WPB;
    const long w0         = (long)(blockIdx.x - NODE_BLOCKS) * WPB + wid;

    for (long tile = w0; tile < full_tiles; tile += stride) {
      const float* p = edge + (tile * 16 + l16) * 5;
      v2f a, a4;
      a4.y = 0.f;
      if (half == 0) {
        a.x  = __builtin_nontemporal_load(p + 0);   // K0
        a.y  = __builtin_nontemporal_load(p + 1);   // K1
        a4.x = __builtin_nontemporal_load(p + 4);   // K=4 feature
      } else {
        a.x  = __builtin_nontemporal_load(p + 2);   // K2
        a.y  = __builtin_nontemporal_load(p + 3);   // K3
        a4.x = 0.f;
      }
      #pragma unroll
      for (int t = 0; t < 4; ++t) {
        v8f d = wmma4(a4, B4[t], cb[t]);   // f4 * W[4] + bias
        d     = wmma4(a,  B03[t], d);      // f0..f3 * W[0..3]
        #pragma unroll
        for (int r = 0; r < 8; ++r) acc[t][r] += relu1(d[r]);
      }
    }
    // masked tail tile (at most one per grid)
    if ((edge_rows & 15) && (full_tiles % stride) == w0) {
      const long tile = full_tiles;
      const long row  = tile * 16 + l16;
      v2f a, a4;
      a.x = 0.f; a.y = 0.f; a4.x = 0.f; a4.y = 0.f;
      if (row < edge_rows) {
        const float* p = edge + row * 5;
        if (half == 0) { a.x = p[0]; a.y = p[1]; a4.x = p[4]; }
        else           { a.x = p[2]; a.y = p[3]; }
      }
      #pragma unroll
      for (int t = 0; t < 4; ++t) {
        v8f d = wmma4(a4, B4[t], cb[t]);
        d     = wmma4(a,  B03[t], d);
        #pragma unroll
        for (int r = 0; r < 8; ++r) {
          const long m = tile * 16 + r + 8 * half;
          acc[t][r] += (m < edge_rows) ? relu1(d[r]) : 0.f;
        }
      }
    }
    gout = ws + 64;     // edge colsum -> ws[64..127]
  }

  // column-sum reduction: in-lane over 8 rows, then across the two lane halves
  #pragma unroll
  for (int t = 0; t < 4; ++t) {
    float s = 0.f;
    #pragma unroll
    for (int r = 0; r < 8; ++r) s += acc[t][r];
    s += __shfl_xor(s, 16, 32);             // lanes l and l^16 share column N
    if (half == 0) atomicAdd(&bsum[l16 + 16 * t], s);
  }
  __syncthreads();
  if (tid < 64) atomicAdd(&gout[tid], bsum[tid]);
}

__device__ __forceinline__ float sigf(float x) {
  return 1.f / (1.f + __expf(-x));
}

__global__ __launch_bounds__(256) void lstm_kernel(
    const float* __restrict__ hist, int T,
    const float* __restrict__ Wih, const float* __restrict__ Whh,
    const float* __restrict__ bih, const float* __restrict__ bhh,
    float* __restrict__ ws)
{
  __shared__ float hs[64], cs[64], gates[256];
  const int g = threadIdx.x;                 // one gate row per thread
  float wrow[64];
  #pragma unroll
  for (int k = 0; k < 64; ++k) wrow[k] = Whh[g * 64 + k];
  const float wi0 = Wih[g * 3 + 0], wi1 = Wih[g * 3 + 1], wi2 = Wih[g * 3 + 2];
  const float bb = bih[g] + bhh[g];
  if (g < 64) { hs[g] = 0.f; cs[g] = 0.f; }
  __syncthreads();

  for (int t = 0; t < T; ++t) {
    const float x0 = hist[t * 3 + 0], x1 = hist[t * 3 + 1], x2 = hist[t * 3 + 2];
    float a0 = bb + wi0 * x0 + wi1 * x1 + wi2 * x2;
    float a1 = 0.f, a2 = 0.f, a3 = 0.f;     // 4 chains to hide FMA latency
    #pragma unroll
    for (int k = 0; k < 64; k += 4) {
      a0 = fmaf(wrow[k + 0], hs[k + 0], a0);
      a1 = fmaf(wrow[k + 1], hs[k + 1], a1);
      a2 = fmaf(wrow[k + 2], hs[k + 2], a2);
      a3 = fmaf(wrow[k + 3], hs[k + 3], a3);
    }
    gates[g] = (a0 + a1) + (a2 + a3);
    __syncthreads();
    if (g < 64) {
      const float ii = sigf(gates[g]);
      const float ff = sigf(gates[64 + g]);
      const float gg = tanhf(gates[128 + g]);
      const float oo = sigf(gates[192 + g]);
      const float c  = ff * cs[g] + ii * gg;
      cs[g] = c;
      hs[g] = oo * tanhf(c);
    }
    __syncthreads();
  }
  if (g < 64) ws[128 + g] = hs[g];
}

__global__ void prep_kernel(float* __restrict__ ws, long node_rows, long edge_rows,
                            const float* __restrict__ W1, const float* __restrict__ b1,
                            const float* __restrict__ W2, const float* __restrict__ b2)
{
  __shared__ float ctx[192];
  const int j = threadIdx.x;                 // 192 threads
  if (j < 64)       ctx[j] = ws[j] / (float)node_rows;   // hn mean
  else if (j < 128) ctx[j] = ws[j] / (float)edge_rows;   // he mean
  else              ctx[j] = ws[j];                       // z_hist
  __syncthreads();
  if (j < 64) {
    float acc = b1[j];
    #pragma unroll 8
    for (int k = 0; k < 192; ++k) acc = fmaf(ctx[k], W1[k * 64 + j], acc);
    ws[192 + j] = acc;                 // base
    ws[256 + j] = W1[192 * 64 + j];    // wa (cand.x column)
    ws[320 + j] = W1[193 * 64 + j];    // wb (cand.y column)
    ws[384 + j] = W2[j];               // w2
    if (j == 0) ws[448] = b2[0];
  }
}

__global__ __launch_bounds__(256) void score_kernel(
    const int* __restrict__ cand, long P, const int* __restrict__ Nptr,
    const float* __restrict__ ws, float* __restrict__ out)
{
  __shared__ float sh[257];                  // base | wa | wb | w2 | b2
  const int tid = threadIdx.x;
  sh[tid] = ws[192 + tid];
  if (tid == 0) sh[256] = ws[448];
  __syncthreads();

  const float denom = (float)(*Nptr - 1) + 1e-9f;
  const float inv   = 1.f / denom;
  const long  stride = (long)gridDim.x * blockDim.x;
  const v2i* cpair = (const v2i*)cand;
  for (long i = (long)blockIdx.x * blockDim.x + tid; i < P; i += stride) {
    const v2i p = __builtin_nontemporal_load(cpair + i);
    const float c0 = (float)p.x * inv;
    const float c1 = (float)p.y * inv;
    float acc = 0.f;
    #pragma unroll
    for (int j = 0; j < 64; ++j) {
      float tt = fmaf(c1, sh[128 + j], fmaf(c0, sh[64 + j], sh[j]));
      tt = relu1(tt);
      acc = fmaf(tt, sh[192 + j], acc);
    }
    __builtin_nontemporal_store(acc + sh[256], &out[i]);
  }
}

extern "C" void kernel_launch(void* const* d_in, const int* in_sizes, int n_in,
                              void* d_out, int out_size, void* d_ws, size_t ws_size,
                              hipStream_t stream)
{
  (void)n_in; (void)out_size; (void)ws_size;

  const float* node  = (const float*)d_in[0];
  const float* edge  = (const float*)d_in[1];
  const float* hist  = (const float*)d_in[2];
  const int*   cand  = (const int*)d_in[3];
  const int*   Nptr  = (const int*)d_in[4];
  const float* nodeW = (const float*)d_in[5];
  const float* nodeB = (const float*)d_in[6];
  const float* edgeW = (const float*)d_in[7];
  const float* edgeB = (const float*)d_in[8];
  const float* Wih   = (const float*)d_in[9];
  const float* Whh   = (const float*)d_in[10];
  const float* bih   = (const float*)d_in[11];
  const float* bhh   = (const float*)d_in[12];
  const float* W1    = (const float*)d_in[13];
  const float* b1    = (const float*)d_in[14];
  const float* W2    = (const float*)d_in[15];
  const float* b2    = (const float*)d_in[16];

  float* ws  = (float*)d_ws;
  float* out = (float*)d_out;

  const long node_rows = in_sizes[0] / 2;
  const long edge_rows = in_sizes[1] / 5;
  const int  T         = in_sizes[2] / 3;
  const long P         = in_sizes[3] / 2;

  hipLaunchKernelGGL(zero_kernel, dim3(1), dim3(128), 0, stream, ws);
  hipLaunchKernelGGL(encoder_kernel, dim3(ENC_BLOCKS), dim3(256), 0, stream,
                     node, node_rows, edge, edge_rows,
                     nodeW, nodeB, edgeW, edgeB, ws);
  hipLaunchKernelGGL(lstm_kernel, dim3(1), dim3(256), 0, stream,
                     hist, T, Wih, Whh, bih, bhh, ws);
  hipLaunchKernelGGL(prep_kernel, dim3(1), dim3(192), 0, stream,
                     ws, node_rows, edge_rows, W1, b1, W2, b2);
  long sblocks = (P + 1023) / 1024;
  if (sblocks > 2048) sblocks = 2048;
  if (sblocks < 1) sblocks = 1;
  hipLaunchKernelGGL(score_kernel, dim3((int)sblocks), dim3(256), 0, stream,
                     cand, P, Nptr, ws, out);
}